// Checkin2POI_16003048145614
// MI455X (gfx1250) — compile-verified
//
#include <hip/hip_runtime.h>
#include <hip/hip_bf16.h>
#include <math.h>

#define HIDDEN 256
#define HEADS  4
#define DHEAD  64

typedef float v2f __attribute__((ext_vector_type(2)));
typedef float v8f __attribute__((ext_vector_type(8)));

// ---- order-preserving float<->uint encoding for atomic max on scores ----
__device__ __forceinline__ unsigned int enc_f32(float f) {
    unsigned int u = __float_as_uint(f);
    return (u & 0x80000000u) ? ~u : (u | 0x80000000u);
}
__device__ __forceinline__ float dec_f32(unsigned int e) {
    unsigned int u = (e & 0x80000000u) ? (e & 0x7FFFFFFFu) : ~e;
    return __uint_as_float(u);
}

// ---------------- zero-fill scratch (smax/denom/poi_agg) ----------------
__global__ void k_zero(float* __restrict__ p, long long n) {
    long long i = (long long)blockIdx.x * blockDim.x + threadIdx.x;
    if (i < n) p[i] = 0.0f;
}

// ---- Q = S@Wq^T + bq ; qkvec_h = Q_h-folded Wk ; qb_h = Q_h . bk_h ----
__global__ void k_qprep(const float* __restrict__ S,  const float* __restrict__ Wq,
                        const float* __restrict__ bq, const float* __restrict__ Wk,
                        const float* __restrict__ bk,
                        float* __restrict__ Qv, float* __restrict__ qkvec,
                        float* __restrict__ qb) {
    __shared__ float sS[HIDDEN];
    __shared__ float sQ[HIDDEN];
    int t = threadIdx.x;
    sS[t] = S[t];
    __syncthreads();
    float acc = bq[t];
    const float* w = Wq + (long long)t * HIDDEN;
    for (int j = 0; j < HIDDEN; ++j) acc += sS[j] * w[j];
    sQ[t] = acc;
    Qv[t] = acc;
    __syncthreads();
    for (int h = 0; h < HEADS; ++h) {
        float a = 0.0f;
        for (int c = 0; c < DHEAD; ++c)
            a += sQ[h * DHEAD + c] * Wk[(long long)(h * DHEAD + c) * HIDDEN + t];
        qkvec[h * HIDDEN + t] = a;
    }
    if (t < HEADS) {
        float a = 0.0f;
        for (int c = 0; c < DHEAD; ++c) a += sQ[t * DHEAD + c] * bk[t * DHEAD + c];
        qb[t] = a;
    }
}

// ---- scores[n,h] = (x[n].qkvec_h + qb_h)/16 ; fused segment atomic max ----
__global__ void k_scores(const float* __restrict__ x, const int* __restrict__ idx,
                         const float* __restrict__ qkvec, const float* __restrict__ qb,
                         float* __restrict__ sc, unsigned int* __restrict__ smax_bits,
                         int N) {
    __shared__ float sqk[HEADS * HIDDEN];
    __shared__ float sqb[HEADS];
    int t = threadIdx.x;
    for (int i = t; i < HEADS * HIDDEN; i += blockDim.x) sqk[i] = qkvec[i];
    if (t < HEADS) sqb[t] = qb[t];
    __syncthreads();
    int wave = t >> 5, lane = t & 31;
    int n = blockIdx.x * 8 + wave;
    if (n >= N) return;
    const float* xr = x + (long long)n * HIDDEN;
    float xv[8];
#pragma unroll
    for (int i = 0; i < 8; ++i) xv[i] = xr[lane + 32 * i];
#pragma unroll
    for (int h = 0; h < HEADS; ++h) {
        float p = 0.0f;
#pragma unroll
        for (int i = 0; i < 8; ++i) p += xv[i] * sqk[h * HIDDEN + lane + 32 * i];
        for (int off = 16; off > 0; off >>= 1) p += __shfl_down(p, off, 32);
        if (lane == 0) {
            float s = (p + sqb[h]) * (1.0f / 16.0f);   // scale = sqrt(256)
            sc[(long long)n * HEADS + h] = s;
            atomicMax(&smax_bits[(long long)idx[n] * HEADS + h], enc_f32(s));
        }
    }
}

// ---- e = exp(s - smax[idx]) (in place) ; denom += e ----
__global__ void k_exp(const int* __restrict__ idx,
                      const unsigned int* __restrict__ smax_bits,
                      float* __restrict__ sc_e, float* __restrict__ denom, int N) {
    int i = blockIdx.x * blockDim.x + threadIdx.x;
    if (i >= N * HEADS) return;
    int n = i >> 2, h = i & 3;
    int p = idx[n];
    unsigned int mb = smax_bits[(long long)p * HEADS + h];
    float m = (mb == 0u) ? 0.0f : dec_f32(mb);
    float e = expf(sc_e[i] - m);
    sc_e[i] = e;
    atomicAdd(&denom[(long long)p * HEADS + h], e);
}

// ---- alpha = e / (denom[idx] + 1e-16) (in place) ----
__global__ void k_alpha(const int* __restrict__ idx, const float* __restrict__ denom,
                        float* __restrict__ e_alpha, int N) {
    int i = blockIdx.x * blockDim.x + threadIdx.x;
    if (i >= N * HEADS) return;
    int n = i >> 2, h = i & 3;
    e_alpha[i] = e_alpha[i] / (denom[(long long)idx[n] * HEADS + h] + 1e-16f);
}

// ======================================================================
// fp32 WMMA GEMM core, batched-load form: per K-step (K=8, two wmma
// K-chunks) issue ALL 18 fragment loads first (one base pointer each +
// compile-time immediate offsets, max 114704 < 2^23), then run the 16
// WMMAs. 18 in-flight loads + in-order load completion lets the backend
// stagger s_wait_loadcnt so later loads overlap earlier WMMAs, while
// register pressure stays ~110 VGPRs (no spills).
// Fragment layout (v_wmma_f32_16x16x4_f32): lane = {M|N}%16, K pair =
// (lane/16)*2 -> one 8-byte (b64) load per fragment per lane.
// ======================================================================
#define WMMA_F32(a, b, c) \
    __builtin_amdgcn_wmma_f32_16x16x4_f32(false, (a), false, (b), (short)0, (c), false, false)

__device__ __forceinline__ void gemm_half_f32(const float* __restrict__ arow,
                                              const float* __restrict__ B,
                                              int colbase, int m, int kb,
                                              v8f acc[8]) {
    const char* ab = (const char*)(arow + kb);
    const char* bb = (const char*)(B + (long long)(colbase + m) * HIDDEN + kb);
    for (int kc = 0; kc < HIDDEN; kc += 8) {
        int ko = kc * 4;                        // byte offset
        v2f a0 = *(const v2f*)(ab + ko);
        v2f a1 = *(const v2f*)(ab + ko + 16);
        v2f b0[8], b1[8];
#pragma unroll
        for (int cg = 0; cg < 8; ++cg) {
            b0[cg] = *(const v2f*)(bb + ko + cg * 16384);
            b1[cg] = *(const v2f*)(bb + ko + 16 + cg * 16384);
        }
#pragma unroll
        for (int cg = 0; cg < 8; ++cg) acc[cg] = WMMA_F32(a0, b0[cg], acc[cg]);
#pragma unroll
        for (int cg = 0; cg < 8; ++cg) acc[cg] = WMMA_F32(a1, b1[cg], acc[cg]);
    }
}

// ---- V = x@Wv^T + bv (fp32 WMMA) fused with alpha-weighted segment sum ----
__global__ void k_vagg(const float* __restrict__ x, const int* __restrict__ idx,
                       const float* __restrict__ Wv, const float* __restrict__ bv,
                       const float* __restrict__ alpha, float* __restrict__ poi_agg,
                       int nRowBlocks) {
    int wave = threadIdx.x >> 5, lane = threadIdx.x & 31;
    int rb = blockIdx.x * 8 + wave;
    if (rb >= nRowBlocks) return;
    int n0 = rb * 16;
    int m  = lane & 15;
    int kb = (lane >> 4) * 2;
    const float* xrow = x + (long long)(n0 + m) * HIDDEN;
    for (int half = 0; half < 2; ++half) {
        int colbase = half * 128;
        v8f zero = {};
        v8f acc[8];
#pragma unroll
        for (int cg = 0; cg < 8; ++cg) acc[cg] = zero;

        gemm_half_f32(xrow, Wv, colbase, m, kb, acc);

        float bvc[8];
#pragma unroll
        for (int cg = 0; cg < 8; ++cg) bvc[cg] = bv[colbase + cg * 16 + m];
#pragma unroll
        for (int r = 0; r < 8; ++r) {
            int row = n0 + r + (lane >> 4) * 8;            // C layout: M = r (+8 hi lanes)
            long long obase = (long long)idx[row] * HIDDEN;
            float alo = alpha[(long long)row * HEADS + half * 2 + 0];  // h for cg 0..3
            float ahi = alpha[(long long)row * HEADS + half * 2 + 1];  // h for cg 4..7
#pragma unroll
            for (int cg = 0; cg < 8; ++cg) {
                int col  = colbase + cg * 16 + m;
                float al = (cg < 4) ? alo : ahi;
                atomicAdd(&poi_agg[obase + col], al * (acc[cg][r] + bvc[cg]));
            }
        }
    }
}

// ---- O1 = Q + poi_agg ; out = prelu(O1 + relu(O1@Wo^T + bo)), nan->0 ----
__global__ void k_out(const float* __restrict__ poi_agg, const float* __restrict__ Qv,
                      const float* __restrict__ Wo, const float* __restrict__ bo,
                      const float* __restrict__ prelu_a, float* __restrict__ out,
                      int nRowBlocks) {
    int wave = threadIdx.x >> 5, lane = threadIdx.x & 31;
    int rb = blockIdx.x * 8 + wave;
    if (rb >= nRowBlocks) return;
    int p0 = rb * 16;
    int m  = lane & 15;
    int kb = (lane >> 4) * 2;
    float pa = *prelu_a;
    const char* prow = (const char*)(poi_agg + (long long)(p0 + m) * HIDDEN + kb);
    const char* qrow = (const char*)(Qv + kb);

    for (int half = 0; half < 2; ++half) {
        int colbase = half * 128;
        v8f zero = {};
        v8f acc[8];
#pragma unroll
        for (int cg = 0; cg < 8; ++cg) acc[cg] = zero;

        const char* bb = (const char*)(Wo + (long long)(colbase + m) * HIDDEN + kb);
        for (int kc = 0; kc < HIDDEN; kc += 8) {
            int ko = kc * 4;
            v2f p0v = *(const v2f*)(prow + ko);
            v2f p1v = *(const v2f*)(prow + ko + 16);
            v2f q0v = *(const v2f*)(qrow + ko);
            v2f q1v = *(const v2f*)(qrow + ko + 16);
            v2f b0[8], b1[8];
#pragma unroll
            for (int cg = 0; cg < 8; ++cg) {
                b0[cg] = *(const v2f*)(bb + ko + cg * 16384);
                b1[cg] = *(const v2f*)(bb + ko + 16 + cg * 16384);
            }
            v2f a0 = p0v + q0v;            // cheap VALU inside the wait shadow
            v2f a1 = p1v + q1v;
#pragma unroll
            for (int cg = 0; cg < 8; ++cg) acc[cg] = WMMA_F32(a0, b0[cg], acc[cg]);
#pragma unroll
            for (int cg = 0; cg < 8; ++cg) acc[cg] = WMMA_F32(a1, b1[cg], acc[cg]);
        }

        float boc[8], qc[8];
#pragma unroll
        for (int cg = 0; cg < 8; ++cg) {
            boc[cg] = bo[colbase + cg * 16 + m];
            qc[cg]  = Qv[colbase + cg * 16 + m];
        }
#pragma unroll
        for (int r = 0; r < 8; ++r) {
            int row = p0 + r + (lane >> 4) * 8;
#pragma unroll
            for (int cg = 0; cg < 8; ++cg) {
                int col  = colbase + cg * 16 + m;
                float o1 = poi_agg[(long long)row * HIDDEN + col] + qc[cg];
                float t  = acc[cg][r] + boc[cg];
                t = t > 0.0f ? t : 0.0f;
                float o = o1 + t;
                o = (o >= 0.0f) ? o : pa * o;
                if (o != o) o = 0.0f;                      // nan_to_num
                out[(long long)row * HIDDEN + col] = o;
            }
        }
    }
}

extern "C" void kernel_launch(void* const* d_in, const int* in_sizes, int n_in,
                              void* d_out, int out_size, void* d_ws, size_t ws_size,
                              hipStream_t stream) {
    const float* x   = (const float*)d_in[0];
    const int*   idx = (const int*)  d_in[1];
    // d_in[2] = num_pois (device scalar; P derived from out_size instead)
    const float* Wq = (const float*)d_in[3];  const float* bq = (const float*)d_in[4];
    const float* Wk = (const float*)d_in[5];  const float* bk = (const float*)d_in[6];
    const float* Wv = (const float*)d_in[7];  const float* bv = (const float*)d_in[8];
    const float* Wo = (const float*)d_in[9];  const float* bo = (const float*)d_in[10];
    const float* S  = (const float*)d_in[11];
    const float* pa = (const float*)d_in[12];
    float* out = (float*)d_out;

    int N = in_sizes[0] / HIDDEN;   // 500000
    int P = out_size    / HIDDEN;   // 50000

    // workspace layout (all fp32-sized words)
    char* w = (char*)d_ws;
    unsigned int* smax_bits = (unsigned int*)w;  w += (size_t)P * HEADS  * 4;
    float* denom   = (float*)w;                  w += (size_t)P * HEADS  * 4;
    float* poi_agg = (float*)w;                  w += (size_t)P * HIDDEN * 4;
    float* e       = (float*)w;                  w += (size_t)N * HEADS  * 4;  // scores -> e -> alpha
    float* Qv      = (float*)w;                  w += (size_t)HIDDEN * 4;
    float* qkvec   = (float*)w;                  w += (size_t)HEADS * HIDDEN * 4;
    float* qb      = (float*)w;                  w += (size_t)HEADS * 4;
    (void)ws_size; (void)n_in;

    long long zn = (long long)P * (HEADS + HEADS + HIDDEN);
    k_zero<<<(unsigned)((zn + 255) / 256), 256, 0, stream>>>((float*)d_ws, zn);

    k_qprep<<<1, 256, 0, stream>>>(S, Wq, bq, Wk, bk, Qv, qkvec, qb);

    k_scores<<<(unsigned)((N + 7) / 8), 256, 0, stream>>>(x, idx, qkvec, qb, e, smax_bits, N);

    unsigned gNH = (unsigned)(((long long)N * HEADS + 255) / 256);
    k_exp  <<<gNH, 256, 0, stream>>>(idx, smax_bits, e, denom, N);
    k_alpha<<<gNH, 256, 0, stream>>>(idx, denom, e, N);

    int rbV = N / 16;   // N divisible by 16 for this problem
    k_vagg<<<(unsigned)((rbV + 7) / 8), 256, 0, stream>>>(x, idx, Wv, bv, e, poi_agg, rbV);

    int rbO = P / 16;   // P divisible by 16 for this problem
    k_out<<<(unsigned)((rbO + 7) / 8), 256, 0, stream>>>(poi_agg, Qv, Wo, bo, pa, out, rbO);
}